// ProgramLineDecoderModel_90615220011535
// MI455X (gfx1250) — compile-verified
//
#include <hip/hip_runtime.h>
#include <hip/hip_bf16.h>

#define T_DIM 128
#define N_DIM 64
#define S_DIM 128
#define D_DIM 512
#define NTYPES 6
#define NDIRS 5

typedef __attribute__((ext_vector_type(2))) float v2f;
typedef __attribute__((ext_vector_type(8))) float v8f;

__device__ __forceinline__ v8f wmma_f32(v2f a, v2f b, v8f c) {
    // D = A(16x4 f32) * B(4x16 f32) + C(16x16 f32); exact fp32 FMA path on CDNA5
    return __builtin_amdgcn_wmma_f32_16x16x4_f32(false, a, false, b, (short)0, c,
                                                 false, false);
}

__device__ __forceinline__ v2f ldf2(const float* p) {
    const float2 t = *reinterpret_cast<const float2*>(p);
    v2f r; r.x = t.x; r.y = t.y;
    return r;
}

// ---------------------------------------------------------------------------
// type_selections[t,n,:] = decoded[t,n,:] @ W_type(512x6) + b_type
// one wave per (t,n) row; lane-strided dot, shfl_xor reduce.  (50 MFLOP)
// ---------------------------------------------------------------------------
__global__ void k_type(const float* __restrict__ dec, const float* __restrict__ Wt,
                       const float* __restrict__ bt, float* __restrict__ out) {
    const int wave = (blockIdx.x * blockDim.x + threadIdx.x) >> 5;
    const int lane = threadIdx.x & 31;
    const float* row = dec + (size_t)wave * D_DIM;
    float acc[NTYPES] = {0.f, 0.f, 0.f, 0.f, 0.f, 0.f};
    for (int k = lane; k < D_DIM; k += 32) {
        const float v = row[k];
#pragma unroll
        for (int j = 0; j < NTYPES; ++j) acc[j] += v * Wt[k * NTYPES + j];
    }
#pragma unroll
    for (int j = 0; j < NTYPES; ++j)
        for (int o = 16; o; o >>= 1) acc[j] += __shfl_xor(acc[j], o, 32);
    if (lane == 0) {
#pragma unroll
        for (int j = 0; j < NTYPES; ++j) out[wave * NTYPES + j] = acc[j] + bt[j];
    }
}

// ---------------------------------------------------------------------------
// pointer_embedding = [decoded | emb[tgt0'] | src_e[tgt1']] @ W_obj + b_obj
// M = 8192, K = 3*512 (gathers folded into per-segment base pointers), N = 512.
// One wave computes a 64x16 strip: 4 sub-tiles share each (strided) W_obj
// B-fragment, cutting W traffic 4x.  fp32 WMMA, K-step 4.
// ---------------------------------------------------------------------------
__global__ void k_pe(const float* __restrict__ dec, const int* __restrict__ tgt,
                     const float* __restrict__ src_e, const float* __restrict__ emb,
                     const float* __restrict__ Wobj, const float* __restrict__ bobj,
                     float* __restrict__ pe) {
    const int lane   = threadIdx.x & 31;
    const int wave   = blockIdx.x * (blockDim.x >> 5) + (threadIdx.x >> 5);
    const int mSuper = wave >> 5;           // 0..127 (64 rows each)
    const int nTile  = wave & 31;           // 0..31  (16 cols each)
    const int l15    = lane & 15;
    const int off    = (lane >> 4) << 1;    // K-half: 0 or 2

    // Per-sub-tile segment base pointers (A row for this lane = m[sub])
    const float* a0[4];  // decoded segment
    const float* a1[4];  // emb segment
    const float* a2[4];  // src_e segment
#pragma unroll
    for (int sub = 0; sub < 4; ++sub) {
        const int m  = mSuper * 64 + sub * 16 + l15;
        const int t  = m >> 6;
        const int n  = m & (N_DIM - 1);
        const int t1 = (t + 1) & (T_DIM - 1);   // roll(-1)
        const int tb = (t1 * N_DIM + n) * 3;
        const int ty = tgt[tb + 0];
        const int qi = tgt[tb + 1];
        a0[sub] = dec   + (size_t)m * D_DIM;
        a1[sub] = emb   + (size_t)ty * D_DIM;
        a2[sub] = src_e + ((size_t)qi * N_DIM + n) * D_DIM;
    }

    const int colW = nTile * 16 + l15;      // B / C column for this lane
    v8f c[4] = {};
#pragma unroll
    for (int s = 0; s < 3; ++s) {
        const float* W = Wobj + (size_t)s * D_DIM * D_DIM + colW;
#pragma unroll 2
        for (int k0 = 0; k0 < D_DIM; k0 += 4) {
            const int kk = k0 + off;
            v2f b;
            b.x = W[(size_t)kk * D_DIM];
            b.y = W[(size_t)(kk + 1) * D_DIM];
#pragma unroll
            for (int sub = 0; sub < 4; ++sub) {
                const float* A = (s == 0) ? a0[sub] : (s == 1) ? a1[sub] : a2[sub];
                c[sub] = wmma_f32(ldf2(A + kk), b, c[sub]);
            }
        }
    }
    const float bias = bobj[colW];
#pragma unroll
    for (int sub = 0; sub < 4; ++sub) {
        const int rbase = mSuper * 64 + sub * 16 + ((lane >> 4) << 3);
#pragma unroll
        for (int r = 0; r < 8; ++r) {
            pe[(size_t)(rbase + r) * D_DIM + colW] = c[sub][r] + bias;
        }
    }
}

// ---------------------------------------------------------------------------
// object_selections[t,n,s] = sum_d src_e[s,n,d] * pe[t,n,d]  (+ mask -> -inf)
// Batched over n.  One wave computes a 64x16 strip of the TxS tile for one n:
// the src_e B-fragment is reused by 4 WMMAs.
// ---------------------------------------------------------------------------
__global__ void k_obj(const float* __restrict__ pe, const float* __restrict__ src_e,
                      const unsigned char* __restrict__ mask, float* __restrict__ out) {
    const int lane = threadIdx.x & 31;
    const int wave = blockIdx.x * (blockDim.x >> 5) + (threadIdx.x >> 5);
    const int sT = wave & 7;                // 0..7  (16 cols of S)
    const int mS = (wave >> 3) & 1;         // 0..1  (64 rows of T)
    const int n  = wave >> 4;               // 0..63
    const int l15 = lane & 15;
    const int off = (lane >> 4) << 1;
    const int colS = sT * 16 + l15;

    const float* A[4];
#pragma unroll
    for (int sub = 0; sub < 4; ++sub) {
        const int rowT = mS * 64 + sub * 16 + l15;
        A[sub] = pe + ((size_t)rowT * N_DIM + n) * D_DIM;
    }
    const float* B = src_e + ((size_t)colS * N_DIM + n) * D_DIM;

    v8f c[4] = {};
#pragma unroll 2
    for (int k0 = 0; k0 < D_DIM; k0 += 4) {
        const int kk = k0 + off;
        const v2f b = ldf2(B + kk);
#pragma unroll
        for (int sub = 0; sub < 4; ++sub) {
            c[sub] = wmma_f32(ldf2(A[sub] + kk), b, c[sub]);
        }
    }
    const bool masked = mask[n * S_DIM + colS] != 0;
    const float neg_inf = -__builtin_inff();
#pragma unroll
    for (int sub = 0; sub < 4; ++sub) {
        const int tbase = mS * 64 + sub * 16 + ((lane >> 4) << 3);
#pragma unroll
        for (int r = 0; r < 8; ++r) {
            const int t = tbase + r;
            out[((size_t)t * N_DIM + n) * S_DIM + colS] = masked ? neg_inf : c[sub][r];
        }
    }
}

// ---------------------------------------------------------------------------
// direction_selections = [decoded | emb[tgt0'] | src_e[tgt1'] | src_e[tgt2']]
//                        @ W_dir(2048x5) + b_dir.  Wave per row.  (167 MFLOP)
// ---------------------------------------------------------------------------
__global__ void k_dir(const float* __restrict__ dec, const int* __restrict__ tgt,
                      const float* __restrict__ src_e, const float* __restrict__ emb,
                      const float* __restrict__ Wd, const float* __restrict__ bd,
                      float* __restrict__ out) {
    const int wave = (blockIdx.x * blockDim.x + threadIdx.x) >> 5;
    const int lane = threadIdx.x & 31;
    const int t  = wave >> 6;
    const int n  = wave & (N_DIM - 1);
    const int t1 = (t + 1) & (T_DIM - 1);
    const int tb = (t1 * N_DIM + n) * 3;
    const int ty = tgt[tb + 0];
    const int qi = tgt[tb + 1];
    const int ri = tgt[tb + 2];

    const float* s0 = dec   + (size_t)wave * D_DIM;
    const float* s1 = emb   + (size_t)ty * D_DIM;
    const float* s2 = src_e + ((size_t)qi * N_DIM + n) * D_DIM;
    const float* s3 = src_e + ((size_t)ri * N_DIM + n) * D_DIM;

    float acc[NDIRS] = {0.f, 0.f, 0.f, 0.f, 0.f};
#pragma unroll
    for (int s = 0; s < 4; ++s) {
        const float* A = (s == 0) ? s0 : (s == 1) ? s1 : (s == 2) ? s2 : s3;
        const float* W = Wd + (size_t)s * D_DIM * NDIRS;
        for (int k = lane; k < D_DIM; k += 32) {
            const float v = A[k];
#pragma unroll
            for (int j = 0; j < NDIRS; ++j) acc[j] += v * W[k * NDIRS + j];
        }
    }
#pragma unroll
    for (int j = 0; j < NDIRS; ++j)
        for (int o = 16; o; o >>= 1) acc[j] += __shfl_xor(acc[j], o, 32);
    if (lane == 0) {
#pragma unroll
        for (int j = 0; j < NDIRS; ++j) out[wave * NDIRS + j] = acc[j] + bd[j];
    }
}

extern "C" void kernel_launch(void* const* d_in, const int* in_sizes, int n_in,
                              void* d_out, int out_size, void* d_ws, size_t ws_size,
                              hipStream_t stream) {
    (void)in_sizes; (void)n_in; (void)out_size; (void)ws_size;
    const float* dec  = (const float*)d_in[0];
    const int*   tgt  = (const int*)d_in[1];
    const float* srcE = (const float*)d_in[2];
    const unsigned char* mask = (const unsigned char*)d_in[3];
    const float* emb  = (const float*)d_in[4];
    const float* Wt   = (const float*)d_in[5];
    const float* bt   = (const float*)d_in[6];
    const float* Wobj = (const float*)d_in[7];
    const float* bobj = (const float*)d_in[8];
    const float* Wd   = (const float*)d_in[9];
    const float* bd   = (const float*)d_in[10];

    float* out      = (float*)d_out;
    float* out_type = out;                                      // T*N*6
    float* out_obj  = out + (size_t)T_DIM * N_DIM * NTYPES;     // T*N*S
    float* out_dir  = out_obj + (size_t)T_DIM * N_DIM * S_DIM;  // T*N*5
    float* pe       = (float*)d_ws;                             // T*N*D f32 (16 MB)

    const int rows = T_DIM * N_DIM;                 // 8192
    // 8 waves / 256-thread block
    k_type<<<rows / 8, 256, 0, stream>>>(dec, Wt, bt, out_type);
    // 128 mSupers * 32 nTiles waves, 4 waves / 128-thread block
    k_pe<<<(128 * 32) / 4, 128, 0, stream>>>(dec, tgt, srcE, emb, Wobj, bobj, pe);
    // 64 n * 2 mSupers * 8 sTiles waves, 4 waves / block (ordered after k_pe)
    k_obj<<<(64 * 2 * 8) / 4, 128, 0, stream>>>(pe, srcE, mask, out_obj);
    k_dir<<<rows / 8, 256, 0, stream>>>(dec, tgt, srcE, emb, Wd, bd, out_dir);
}